// MGDCF_38800734552802
// MI455X (gfx1250) — compile-verified
//
#include <hip/hip_runtime.h>

// ---------------------------------------------------------------------------
// MGDCF diffusion on MI455X (gfx1250, wave32).
//
//   1) norm[n] = 1                       (self loop degree)
//   2) norm[u]+=1, norm[U+i]+=1          (atomic degree accumulation per edge)
//   3) norm[n] = rsqrt(norm[n])
//   4) 4x { dense init pass, edge scatter pass }   ping-pong hbuf <-> d_out
//
// Per step:  hout[n,:] = s*(BETA*norm[n]^2*hin[n,:] + ALPHA*x[n,:])
//            per UI edge (u,v): w = s*BETA*norm[u]*norm[v]
//                 hout[v,:] += w*hin[u,:];  hout[u,:] += w*hin[v,:]
//
// CDNA5 specifics:
//  * GLOBAL_LOAD_ASYNC_TO_LDS_B128 gathers both 256B rows of an edge in ONE
//    instruction (16B/lane; lanes 0-15 -> row u, lanes 16-31 -> row v),
//    double-buffered per wave, tracked with ASYNCcnt (s_wait_asynccnt).
//  * Non-returning global_atomic_add_f32 at DEV scope for the scatter-add
//    (emitted directly via asm to guarantee single-instruction fp atomics).
//  * h buffers (2 x 76.8 MB) are L2-resident (192 MB L2) -> ~5 GB traffic,
//    ~220 us roofline at 23.3 TB/s.
// ---------------------------------------------------------------------------

namespace {
constexpr int NUSERS = 200000;
constexpr int NITEMS = 100000;
constexpr int NNODES = NUSERS + NITEMS;   // 300000
constexpr int NEDGES = 1000000;
constexpr int D      = 64;
constexpr float ALPHA = 0.1f;
constexpr float BETA  = 0.9f;
constexpr int EDGES_PER_WAVE = 32;        // one wave32 handles 32 edges
constexpr int WAVES_PER_BLOCK = 8;        // 256 threads
}

// Non-returning fp32 atomic add, device scope (visible across WGPs).
// Tracked by STOREcnt; implicitly drained by s_endpgm.
__device__ __forceinline__ void atomic_add_f32_dev(float* p, float v) {
  asm volatile("global_atomic_add_f32 %0, %1, off scope:SCOPE_DEV"
               :: "v"((unsigned long long)(uintptr_t)p), "v"(v)
               : "memory");
}

// Per-lane async gather: 16 bytes from global address -> LDS byte offset.
// (VDST = LDS offset VGPR, VADDR = 64-bit global address, saddr = off/NULL)
__device__ __forceinline__ void async_gather16(const float* g, unsigned loff) {
  asm volatile("global_load_async_to_lds_b128 %0, %1, off"
               :: "v"(loff), "v"((unsigned long long)(uintptr_t)g)
               : "memory");
}
__device__ __forceinline__ void wait_async_le1() {
  asm volatile("s_wait_asynccnt 0x1" ::: "memory");
}
__device__ __forceinline__ void wait_async_0() {
  asm volatile("s_wait_asynccnt 0x0" ::: "memory");
}

// Edge pass with CDNA5 async-to-LDS double buffering. (First in file so the
// disasm snippet shows the hot kernel.)
__global__ void __launch_bounds__(256)
k_edge_scatter(const int* __restrict__ uidx, const int* __restrict__ iidx,
               const float* __restrict__ hin, const float* __restrict__ norm,
               float* __restrict__ hout, float scale) {
  // [wave][slot][row u floats 0..63 | row v floats 64..127]
  __shared__ float smem[WAVES_PER_BLOCK][2][2 * D];

  const int lane = threadIdx.x & 31;
  const int wv   = threadIdx.x >> 5;
  const int wave = (blockIdx.x * blockDim.x + threadIdx.x) >> 5;
  if (wave >= NEDGES / EDGES_PER_WAVE) return;

  const int e   = wave * EDGES_PER_WAVE + lane;   // NEDGES % 32 == 0
  const int myU = uidx[e];                        // coalesced index loads
  const int myV = NUSERS + iidx[e];

  // LDS byte offsets of this wave's two slots, per lane (16 B per lane).
  // Low 32 bits of a generic LDS address are the LDS byte offset.
  const unsigned slot0 = (unsigned)(uintptr_t)&smem[wv][0][0] + (unsigned)lane * 16u;
  const unsigned slot1 = slot0 + (unsigned)(2 * D * sizeof(float));

  const float sb = scale * BETA;
  const int d0 = lane * 2;                        // this lane's feature dims

  // ---- issue edge 0's gather: lanes 0-15 cover row u, 16-31 cover row v ----
  {
    const int u0 = __builtin_amdgcn_readlane(myU, 0);
    const int v0 = __builtin_amdgcn_readlane(myV, 0);
    const int row = (lane < 16) ? u0 : v0;
    async_gather16(hin + (size_t)row * D + (lane & 15) * 4, slot0);
  }

#pragma unroll 2
  for (int j = 0; j < EDGES_PER_WAVE - 1; ++j) {
    // issue edge j+1 into the other slot while edge j is consumed
    {
      const int un = __builtin_amdgcn_readlane(myU, j + 1);
      const int vn = __builtin_amdgcn_readlane(myV, j + 1);
      const int row = (lane < 16) ? un : vn;
      const unsigned dst = ((j + 1) & 1) ? slot1 : slot0;
      async_gather16(hin + (size_t)row * D + (lane & 15) * 4, dst);
    }
    wait_async_le1();                 // async completes in order: edge j ready

    const int u = __builtin_amdgcn_readlane(myU, j);
    const int v = __builtin_amdgcn_readlane(myV, j);
    const float w = sb * norm[u] * norm[v];       // scalar loads
    const float* s = &smem[wv][j & 1][0];
    const float2 hu = *(const float2*)(s + d0);
    const float2 hv = *(const float2*)(s + D + d0);
    float* pu = hout + (size_t)u * D + d0;
    float* pv = hout + (size_t)v * D + d0;
    atomic_add_f32_dev(pv + 0, w * hu.x);         // msg user -> item
    atomic_add_f32_dev(pv + 1, w * hu.y);
    atomic_add_f32_dev(pu + 0, w * hv.x);         // msg item -> user
    atomic_add_f32_dev(pu + 1, w * hv.y);
  }

  // ---- drain: last edge ----
  wait_async_0();
  {
    const int j = EDGES_PER_WAVE - 1;
    const int u = __builtin_amdgcn_readlane(myU, j);
    const int v = __builtin_amdgcn_readlane(myV, j);
    const float w = sb * norm[u] * norm[v];
    const float* s = &smem[wv][j & 1][0];
    const float2 hu = *(const float2*)(s + d0);
    const float2 hv = *(const float2*)(s + D + d0);
    float* pu = hout + (size_t)u * D + d0;
    float* pv = hout + (size_t)v * D + d0;
    atomic_add_f32_dev(pv + 0, w * hu.x);
    atomic_add_f32_dev(pv + 1, w * hu.y);
    atomic_add_f32_dev(pu + 0, w * hv.x);
    atomic_add_f32_dev(pu + 1, w * hv.y);
  }
}

__global__ void k_deg_init(float* __restrict__ deg) {
  int i = blockIdx.x * blockDim.x + threadIdx.x;
  if (i < NNODES) deg[i] = 1.0f;          // self loop
}

__global__ void k_deg_accum(const int* __restrict__ uidx,
                            const int* __restrict__ iidx,
                            float* __restrict__ deg) {
  int e = blockIdx.x * blockDim.x + threadIdx.x;
  if (e < NEDGES) {
    atomic_add_f32_dev(&deg[uidx[e]], 1.0f);
    atomic_add_f32_dev(&deg[NUSERS + iidx[e]], 1.0f);
  }
}

__global__ void k_norm_finalize(float* __restrict__ deg) {
  int i = blockIdx.x * blockDim.x + threadIdx.x;
  if (i < NNODES) deg[i] = rsqrtf(deg[i]);
}

// Dense pass: hout[n,:] = s*(BETA*norm[n]^2*hin[n,:] + ALPHA*x0[n,:])
__global__ void __launch_bounds__(256)
k_step_init(const float4* __restrict__ hin, const float4* __restrict__ x0,
            const float* __restrict__ norm, float4* __restrict__ hout,
            float scale) {
  int i = blockIdx.x * blockDim.x + threadIdx.x;
  if (i >= NNODES * (D / 4)) return;
  int n = i / (D / 4);
  float nr = norm[n];
  float a = scale * BETA * nr * nr;       // self-loop weight * beta * s
  float b = scale * ALPHA;
  float4 h = hin[i];
  float4 h0 = x0[i];
  float4 o;
  o.x = a * h.x + b * h0.x;
  o.y = a * h.y + b * h0.y;
  o.z = a * h.z + b * h0.z;
  o.w = a * h.w + b * h0.w;
  hout[i] = o;
}

extern "C" void kernel_launch(void* const* d_in, const int* in_sizes, int n_in,
                              void* d_out, int out_size, void* d_ws, size_t ws_size,
                              hipStream_t stream) {
  const float* x    = (const float*)d_in[0];   // [N, 64] fp32
  const int*   uidx = (const int*)d_in[1];     // [1M]
  const int*   iidx = (const int*)d_in[2];     // [1M]
  float* out  = (float*)d_out;                 // [N, 64] fp32
  float* hbuf = (float*)d_ws;                  // N*D floats   (76.8 MB)
  float* norm = hbuf + (size_t)NNODES * D;     // N floats     (1.2 MB)

  auto cdiv = [](int a, int b) { return (a + b - 1) / b; };
  const int T = 256;   // 8 waves per block on wave32

  // --- degree / normalization ---
  k_deg_init<<<cdiv(NNODES, T), T, 0, stream>>>(norm);
  k_deg_accum<<<cdiv(NEDGES, T), T, 0, stream>>>(uidx, iidx, norm);
  k_norm_finalize<<<cdiv(NNODES, T), T, 0, stream>>>(norm);

  // GAMMA = BETA^K + ALPHA * sum_{i<K} BETA^i   (== 1.0 for these constants)
  double bk = 1.0, ssum = 0.0;
  for (int i = 0; i < 4; ++i) { ssum += bk; bk *= (double)BETA; }
  const float inv_gamma = (float)(1.0 / (bk + (double)ALPHA * ssum));

  const int init_blocks = cdiv(NNODES * (D / 4), T);
  const int edge_blocks = cdiv(NEDGES, T);     // one lane per edge

  // --- K = 4 diffusion steps, ping-pong: hbuf -> out -> hbuf -> out ---
  const float* hin = x;
  float* bufs[4] = { hbuf, out, hbuf, out };
  for (int step = 0; step < 4; ++step) {
    float* hout = bufs[step];
    const float s = (step == 3) ? inv_gamma : 1.0f;
    k_step_init<<<init_blocks, T, 0, stream>>>(
        (const float4*)hin, (const float4*)x, norm, (float4*)hout, s);
    k_edge_scatter<<<edge_blocks, T, 0, stream>>>(
        uidx, iidx, hin, norm, hout, s);
    hin = hout;
  }
}